// EdgeClassifier_82798379532714
// MI455X (gfx1250) — compile-verified
//
#include <hip/hip_runtime.h>
#include <hip/hip_bf16.h>

// ---------------------------------------------------------------------------
// EdgeClassifier for MI455X (gfx1250, wave32).
//   1) degree (f32 atomics) + deg_inv
//   2) scatter-sum x[src] -> agg[dst]
//   3) h1 = relu([agg*dinv | x] @ [W1l;W1r] + b1l)   V_WMMA_F32_16X16X4_F32
//   4) scatter-sum h1[src] -> agg[dst]
//   5) h2 = relu([agg*dinv | h1] @ [W2l;W2r] + b2l)
//   6) [h2[src] | h2[dst] | edge_attr] (16x144) @ Wm1, ReLU, dot Wm2
// WMMA operands: A preloaded into registers per K-segment (no branches in the
// MAC loop); B pre-swizzled in LDS to the per-lane v2f layout (one ds_load_b64
// per WMMA, conflict-free).
// ---------------------------------------------------------------------------

typedef __attribute__((ext_vector_type(2))) float v2f;
typedef __attribute__((ext_vector_type(8))) float v8f;

#define N_NODES 50000
#define N_EDGES 800000

__device__ __forceinline__ v8f wmma4(v2f a, v2f b, v8f c) {
  return __builtin_amdgcn_wmma_f32_16x16x4_f32(
      false, a, false, b, (short)0, c, false, false);
}

// ----------------------------- utility kernels -----------------------------

__global__ void zero_f32(float* __restrict__ p, int n) {
  int i = blockIdx.x * blockDim.x + threadIdx.x;
  if (i < n) p[i] = 0.0f;
}

__global__ void deg_kernel(const long long* __restrict__ dst,
                           float* __restrict__ degf, int E) {
  int e = blockIdx.x * blockDim.x + threadIdx.x;
  if (e < E) atomicAdd(&degf[dst[e]], 1.0f);
}

__global__ void dinv_kernel(const float* __restrict__ degf,
                            float* __restrict__ dinv, int n) {
  int i = blockIdx.x * blockDim.x + threadIdx.x;
  if (i < n) dinv[i] = 1.0f / fmaxf(degf[i], 1.0f);
}

// scatter-sum of 64-wide features: one thread = 4 channels of one edge
__global__ void scatter64_kernel(const float* __restrict__ feat,
                                 const long long* __restrict__ src,
                                 const long long* __restrict__ dst,
                                 float* __restrict__ agg, int E) {
  int t = blockIdx.x * blockDim.x + threadIdx.x;
  if (t >= E * 16) return;
  int e  = t >> 4;
  int c4 = (t & 15) << 2;
  long long s = src[e], d = dst[e];
  const float4 v = *(const float4*)(feat + (long long)s * 64 + c4);
  float* ap = agg + (long long)d * 64 + c4;
  atomicAdd(ap + 0, v.x);
  atomicAdd(ap + 1, v.y);
  atomicAdd(ap + 2, v.z);
  atomicAdd(ap + 3, v.w);
}

// --------------------------- WMMA node GEMM --------------------------------
// One wave per 16-node tile: [agg*dinv | xin](16x128) @ [Wl;Wr](128x64) + bl.

#define NK_NODE 32  // K-steps of 4 (K = 128)

__global__ __launch_bounds__(256) void node_gemm_kernel(
    const float* __restrict__ agg, const float* __restrict__ dinv,
    const float* __restrict__ xin,
    const float* __restrict__ Wl, const float* __restrict__ bl,
    const float* __restrict__ Wr,
    float* __restrict__ hout, int nTiles) {
  // B pre-swizzled: Blds[ks*128 + t*32 + lane] = {W[4ks+2hi][n], W[4ks+2hi+1][n]}
  // with n = (lane&15) + 16t, hi = lane>>4.
  __shared__ v2f Blds[NK_NODE * 4 * 32];
  for (int j = threadIdx.x; j < NK_NODE * 4 * 32; j += 256) {
    int ks = j >> 7;
    int t  = (j >> 5) & 3;
    int ln = j & 31;
    int r  = ks * 4 + ((ln >> 4) << 1);         // even row in [0,128)
    int n  = (ln & 15) + 16 * t;
    const float* Wsel = (r < 64) ? (Wl + r * 64) : (Wr + (r - 64) * 64);
    v2f b; b.x = Wsel[n]; b.y = Wsel[64 + n];
    Blds[j] = b;
  }
  __syncthreads();

  const int wave = threadIdx.x >> 5;
  const int lane = threadIdx.x & 31;
  const int tile = blockIdx.x * 8 + wave;
  if (tile >= nTiles) return;  // wave-uniform

  const int m  = lane & 15;
  const int kb = (lane >> 4) << 1;
  const int node = tile * 16 + m;
  const float di = dinv[node];
  const float* __restrict__ arow = agg + (long long)node * 64;
  const float* __restrict__ xrow = xin + (long long)node * 64;

  v8f c0 = {}, c1 = {}, c2 = {}, c3 = {};
  v2f aReg[16];

  // --- segment 0: scaled aggregation (K steps 0..15) ---
  #pragma unroll
  for (int i = 0; i < 16; ++i) {
    v2f t = *(const v2f*)(arow + i * 4 + kb);
    aReg[i].x = t.x * di; aReg[i].y = t.y * di;
  }
  #pragma unroll
  for (int ks = 0; ks < 16; ++ks) {
    const v2f* bp = &Blds[ks * 128 + lane];
    c0 = wmma4(aReg[ks], bp[0],  c0);
    c1 = wmma4(aReg[ks], bp[32], c1);
    c2 = wmma4(aReg[ks], bp[64], c2);
    c3 = wmma4(aReg[ks], bp[96], c3);
  }
  // --- segment 1: root features (K steps 16..31) ---
  #pragma unroll
  for (int i = 0; i < 16; ++i) aReg[i] = *(const v2f*)(xrow + i * 4 + kb);
  #pragma unroll
  for (int ks = 0; ks < 16; ++ks) {
    const v2f* bp = &Blds[(16 + ks) * 128 + lane];
    c0 = wmma4(aReg[ks], bp[0],  c0);
    c1 = wmma4(aReg[ks], bp[32], c1);
    c2 = wmma4(aReg[ks], bp[64], c2);
    c3 = wmma4(aReg[ks], bp[96], c3);
  }

  const float bb0 = bl[m], bb1 = bl[m + 16], bb2 = bl[m + 32], bb3 = bl[m + 48];
  const int hi = lane >> 4;
  float* __restrict__ ob = hout + (long long)(tile * 16) * 64;
  #pragma unroll
  for (int r = 0; r < 8; ++r) {
    const int row = r + 8 * hi;
    ob[row * 64 + m]      = fmaxf(c0[r] + bb0, 0.0f);
    ob[row * 64 + m + 16] = fmaxf(c1[r] + bb1, 0.0f);
    ob[row * 64 + m + 32] = fmaxf(c2[r] + bb2, 0.0f);
    ob[row * 64 + m + 48] = fmaxf(c3[r] + bb3, 0.0f);
  }
}

// ----------------------------- WMMA edge MLP -------------------------------

#define NK_EDGE 36  // K-steps of 4 (K = 144)

__global__ __launch_bounds__(256) void edge_mlp_kernel(
    const float* __restrict__ h,
    const long long* __restrict__ src, const long long* __restrict__ dst,
    const float* __restrict__ ea,
    const float* __restrict__ Wm1, const float* __restrict__ bm1,
    const float* __restrict__ Wm2, const float* __restrict__ bm2,
    float* __restrict__ out, int nTiles) {
  __shared__ v2f Blds[NK_EDGE * 4 * 32];   // 36 KB, pre-swizzled Wm1
  for (int j = threadIdx.x; j < NK_EDGE * 4 * 32; j += 256) {
    int ks = j >> 7;
    int t  = (j >> 5) & 3;
    int ln = j & 31;
    int r  = ks * 4 + ((ln >> 4) << 1);
    int n  = (ln & 15) + 16 * t;
    v2f b; b.x = Wm1[r * 64 + n]; b.y = Wm1[(r + 1) * 64 + n];
    Blds[j] = b;
  }
  __syncthreads();

  const int wave = threadIdx.x >> 5;
  const int lane = threadIdx.x & 31;
  const int tile = blockIdx.x * 8 + wave;
  if (tile >= nTiles) return;

  const int m  = lane & 15;
  const int hi = lane >> 4;
  const int kb = hi << 1;
  const long long e0 = (long long)tile * 16;
  const long long s = src[e0 + m], d = dst[e0 + m];
  const float* __restrict__ hs = h + s * 64;      // hot in 192 MB L2
  const float* __restrict__ hd = h + d * 64;
  const float* __restrict__ er = ea + (e0 + m) * 16;

  v8f c0 = {}, c1 = {}, c2 = {}, c3 = {};
  v2f aReg[16];

  // --- segment 0: h[src] (K steps 0..15) ---
  #pragma unroll
  for (int i = 0; i < 16; ++i) aReg[i] = *(const v2f*)(hs + i * 4 + kb);
  #pragma unroll
  for (int ks = 0; ks < 16; ++ks) {
    const v2f* bp = &Blds[ks * 128 + lane];
    c0 = wmma4(aReg[ks], bp[0],  c0);
    c1 = wmma4(aReg[ks], bp[32], c1);
    c2 = wmma4(aReg[ks], bp[64], c2);
    c3 = wmma4(aReg[ks], bp[96], c3);
  }
  // --- segment 1: h[dst] (K steps 16..31) ---
  #pragma unroll
  for (int i = 0; i < 16; ++i) aReg[i] = *(const v2f*)(hd + i * 4 + kb);
  #pragma unroll
  for (int ks = 0; ks < 16; ++ks) {
    const v2f* bp = &Blds[(16 + ks) * 128 + lane];
    c0 = wmma4(aReg[ks], bp[0],  c0);
    c1 = wmma4(aReg[ks], bp[32], c1);
    c2 = wmma4(aReg[ks], bp[64], c2);
    c3 = wmma4(aReg[ks], bp[96], c3);
  }
  // --- segment 2: edge_attr (K steps 32..35) ---
  #pragma unroll
  for (int i = 0; i < 4; ++i) aReg[i] = *(const v2f*)(er + i * 4 + kb);
  #pragma unroll
  for (int ks = 0; ks < 4; ++ks) {
    const v2f* bp = &Blds[(32 + ks) * 128 + lane];
    c0 = wmma4(aReg[ks], bp[0],  c0);
    c1 = wmma4(aReg[ks], bp[32], c1);
    c2 = wmma4(aReg[ks], bp[64], c2);
    c3 = wmma4(aReg[ks], bp[96], c3);
  }

  // out[row] = sum_n relu(hidden[row][n] + bm1[n]) * Wm2[n] + bm2
  const float w20 = Wm2[m],      w21 = Wm2[m + 16];
  const float w22 = Wm2[m + 32], w23 = Wm2[m + 48];
  const float bb0 = bm1[m],      bb1 = bm1[m + 16];
  const float bb2 = bm1[m + 32], bb3 = bm1[m + 48];
  const float obias = bm2[0];
  #pragma unroll
  for (int r = 0; r < 8; ++r) {
    float p = fmaxf(c0[r] + bb0, 0.0f) * w20
            + fmaxf(c1[r] + bb1, 0.0f) * w21
            + fmaxf(c2[r] + bb2, 0.0f) * w22
            + fmaxf(c3[r] + bb3, 0.0f) * w23;
    p += __shfl_xor(p, 1);
    p += __shfl_xor(p, 2);
    p += __shfl_xor(p, 4);
    p += __shfl_xor(p, 8);   // reduce within each 16-lane half
    if (m == 0) out[e0 + r + 8 * hi] = p + obias;
  }
}

// ------------------------------- launcher ----------------------------------

extern "C" void kernel_launch(void* const* d_in, const int* in_sizes, int n_in,
                              void* d_out, int out_size, void* d_ws, size_t ws_size,
                              hipStream_t stream) {
  (void)in_sizes; (void)n_in; (void)out_size; (void)ws_size;
  const float*     x   = (const float*)d_in[0];
  const long long* ei  = (const long long*)d_in[1];   // jnp.int64 [2, E]
  const float*     ea  = (const float*)d_in[2];
  const float*     W1l = (const float*)d_in[3];
  const float*     b1l = (const float*)d_in[4];
  const float*     W1r = (const float*)d_in[5];
  const float*     W2l = (const float*)d_in[6];
  const float*     b2l = (const float*)d_in[7];
  const float*     W2r = (const float*)d_in[8];
  const float*     Wm1 = (const float*)d_in[9];
  const float*     bm1 = (const float*)d_in[10];
  const float*     Wm2 = (const float*)d_in[11];
  const float*     bm2 = (const float*)d_in[12];
  float* outp = (float*)d_out;

  const long long* src = ei;
  const long long* dst = ei + N_EDGES;

  float* ws    = (float*)d_ws;
  float* degf  = ws;                       // 50048 (padded)
  float* agg   = degf + 50048;             // 3,200,000
  float* dinv  = agg  + 3200000;           // 50048
  float* h1    = dinv + 50048;             // 3,200,000
  float* h2    = h1   + 3200000;           // 3,200,000

  const int TPB = 256;
  const int nNodeTiles = N_NODES / 16;     // 3125
  const int nEdgeTiles = N_EDGES / 16;     // 50000

  {
    int n = 50048 + 3200000;               // degf + agg contiguous
    zero_f32<<<(n + TPB - 1) / TPB, TPB, 0, stream>>>(degf, n);
  }
  deg_kernel<<<(N_EDGES + TPB - 1) / TPB, TPB, 0, stream>>>(dst, degf, N_EDGES);
  dinv_kernel<<<(N_NODES + TPB - 1) / TPB, TPB, 0, stream>>>(degf, dinv, N_NODES);

  scatter64_kernel<<<(N_EDGES * 16 + TPB - 1) / TPB, TPB, 0, stream>>>(
      x, src, dst, agg, N_EDGES);
  node_gemm_kernel<<<(nNodeTiles + 7) / 8, TPB, 0, stream>>>(
      agg, dinv, x, W1l, b1l, W1r, h1, nNodeTiles);

  zero_f32<<<(3200000 + TPB - 1) / TPB, TPB, 0, stream>>>(agg, 3200000);
  scatter64_kernel<<<(N_EDGES * 16 + TPB - 1) / TPB, TPB, 0, stream>>>(
      h1, src, dst, agg, N_EDGES);
  node_gemm_kernel<<<(nNodeTiles + 7) / 8, TPB, 0, stream>>>(
      agg, dinv, h1, W2l, b2l, W2r, h2, nNodeTiles);

  edge_mlp_kernel<<<nEdgeTiles / 8, TPB, 0, stream>>>(
      h2, src, dst, ea, Wm1, bm1, Wm2, bm2, outp, nEdgeTiles);
}